// PScanModel_52553219833872
// MI455X (gfx1250) — compile-verified
//
#include <hip/hip_runtime.h>
#include <cstdint>

// Problem constants (match reference: B=2, L=2048, D=1024, N=16)
#define BATCH 2
#define LDIM  2048
#define DN    16384                 // D*N, contiguous innermost extent per (b,t)
#define TC    32                    // time steps per chunk
#define NBUF  8                     // LDS ring depth
#define NCHUNK (LDIM / TC)          // 64 chunks
#define CPW   32                    // columns per wave (== wave32 lanes)
#define CHUNK_FLOATS (TC * CPW)     // 1024 floats per buffer per array

// One wave32 per block. Each wave owns 32 contiguous (b,dn) columns and runs
// the full sequential scan over L, double(8x)-buffered via async loads to LDS.
__global__ __launch_bounds__(32)
void pscan_async_kernel(const float* __restrict__ A,
                        const float* __restrict__ X,
                        float* __restrict__ H) {
  // 8 ring buffers x 1024 floats per array = 32 KB + 32 KB = 64 KB LDS.
  __shared__ float sA[NBUF * CHUNK_FLOATS];
  __shared__ float sX[NBUF * CHUNK_FLOATS];

  const int lane    = threadIdx.x;                 // 0..31
  const int wave    = blockIdx.x;                  // 0..1023
  const int b       = wave >> 9;                   // 512 waves per batch (16384/32)
  const int colBase = (wave & 511) * CPW;          // dn tile base

  // Async B128 mapping: instruction i covers 4 t-rows; lane -> (t-row r, 4 cols)
  const int r  = lane >> 3;                        // 0..3  sub t-row
  const int c4 = (lane & 7) * 4;                   // 0,4,...,28 first column

  const size_t batchOff = (size_t)b * LDIM * DN;
  const float* gA = A + batchOff + colBase + c4;
  const float* gX = X + batchOff + colBase + c4;

  // LDS byte addresses for the async destination (generic ptr low 32 bits are
  // the LDS offset on gfx1250).
  const unsigned ldsA0 =
      (unsigned)(uintptr_t)sA + (unsigned)((r * CPW + c4) * 4);
  const unsigned ldsX0 =
      (unsigned)(uintptr_t)sX + (unsigned)((r * CPW + c4) * 4);

  auto issue_chunk = [&](int cb) {
    const unsigned bofs = (unsigned)((cb & (NBUF - 1)) * CHUNK_FLOATS * 4);
    unsigned la = ldsA0 + bofs;
    unsigned lx = ldsX0 + bofs;
    const float* pa = gA + (size_t)(cb * TC + r) * DN;
    const float* px = gX + (size_t)(cb * TC + r) * DN;
#pragma unroll
    for (int i = 0; i < TC / 4; ++i) {             // 8 insts per array, 16 total
      // Each lane copies 16 contiguous global bytes (4 cols of one t-row) into
      // the matching [t][col] slot in LDS. Wave-wide: 512 B contiguous in LDS, 4 x 128 B rows DN apart in global.
      asm volatile("global_load_async_to_lds_b128 %0, %1, off"
                   :: "v"(la), "v"(pa) : "memory");
      asm volatile("global_load_async_to_lds_b128 %0, %1, off"
                   :: "v"(lx), "v"(px) : "memory");
      pa += (size_t)4 * DN;
      px += (size_t)4 * DN;
      la += 4 * CPW * 4;
      lx += 4 * CPW * 4;
    }
  };

  // Prologue: 3 chunks in flight (48 async loads <= ASYNCcnt cap 63).
  issue_chunk(0);
  issue_chunk(1);
  issue_chunk(2);

  float h = 0.0f;
  float* outp = H + batchOff + colBase + lane;

  for (int cb = 0; cb < NCHUNK; ++cb) {
    // Async loads complete in order. With buffers cb..cb+2 issued (48 loads),
    // ASYNCcnt <= 32 implies buffer cb's 16 loads are done. Tail chunks issue
    // fewer ahead, so tighten the threshold accordingly.
    if (cb < NCHUNK - 2) {
      asm volatile("s_wait_asynccnt 32" ::: "memory");
    } else if (cb == NCHUNK - 2) {
      asm volatile("s_wait_asynccnt 16" ::: "memory");
    } else {
      asm volatile("s_wait_asynccnt 0" ::: "memory");
    }
    if (cb + 3 < NCHUNK) issue_chunk(cb + 3);      // keep pipeline full

    const float* bA = sA + (cb & (NBUF - 1)) * CHUNK_FLOATS;
    const float* bX = sX + (cb & (NBUF - 1)) * CHUNK_FLOATS;
#pragma unroll
    for (int t = 0; t < TC; ++t) {
      // Lanes read 32 consecutive floats: conflict-free 128 B LDS row.
      float a = bA[t * CPW + lane];
      float x = bX[t * CPW + lane];
      h = __builtin_fmaf(a, h, x);                 // serial carry per column
      *outp = h;                                   // coalesced 128 B wave store
      outp += DN;
    }
  }
}

extern "C" void kernel_launch(void* const* d_in, const int* in_sizes, int n_in,
                              void* d_out, int out_size, void* d_ws, size_t ws_size,
                              hipStream_t stream) {
  (void)in_sizes; (void)n_in; (void)out_size; (void)d_ws; (void)ws_size;
  const float* A = (const float*)d_in[0];
  const float* X = (const float*)d_in[1];
  float*       H = (float*)d_out;
  // 32768 columns / 32 per wave = 1024 single-wave blocks, 64 KB LDS each.
  pscan_async_kernel<<<dim3(1024), dim3(32), 0, stream>>>(A, X, H);
}